// RankNetLoss_3539053052344
// MI455X (gfx1250) — compile-verified
//
#include <hip/hip_runtime.h>
#include <hip/hip_bf16.h>
#include <math.h>

typedef __attribute__((ext_vector_type(2))) float v2f;
typedef __attribute__((ext_vector_type(8))) float v8f;

#define L2E 1.4426950408889634f
#define LN2 0.6931471805599453f

// One 128x128 element block of the (upper-triangular) pair matrix per workgroup.
// 8 waves x 8 subtiles; each 16x16 subtile of d = p_i - p_j is produced by one
// v_wmma_f32_16x16x4_f32 (rank-2 trick: d = [p,1] x [1; -p^T]).
__global__ __launch_bounds__(256) void ranknet_tile_kernel(
    const float* __restrict__ preds, const float* __restrict__ targets,
    double* __restrict__ block_part, int NB)
{
  __shared__ __align__(16) float p_row_s[128];
  __shared__ __align__(16) float t_row_s[128];
  __shared__ __align__(16) float p_col_s[128];
  __shared__ __align__(16) float t_col_s[128];
  __shared__ float wsum_s[8];

  // Map flat block id -> upper-triangle (bi, bj), bi <= bj.
  int rem = (int)blockIdx.x;
  int bi = 0;
  while (rem >= NB - bi) { rem -= NB - bi; ++bi; }
  int bj = bi + rem;

  int t = threadIdx.x;
  if (t < 128) {
    p_row_s[t] = preds[bi * 128 + t];
    t_row_s[t] = targets[bi * 128 + t];
  } else {
    int u = t - 128;
    p_col_s[u] = preds[bj * 128 + u];
    t_col_s[u] = targets[bj * 128 + u];
  }
  __syncthreads();

  int wave = t >> 5;       // 8 waves per block (wave32)
  int lane = t & 31;
  bool lo  = lane < 16;
  int  l15 = lane & 15;

  float acc = 0.0f;

  for (int s = wave; s < 64; s += 8) {
    int wi = s >> 3;
    int wj = s & 7;
    if (bi == bj && wi > wj) continue;   // wave-uniform skip of lower subtiles

    // A (16x4): row i = [p_i, 1, 0, 0]; B (4x16): col j = [1, -p_j, 0, 0]^T
    float pr = p_row_s[wi * 16 + l15];
    float pc = p_col_s[wj * 16 + l15];
    v2f A; A.x = lo ? pr   : 0.0f;  A.y = lo ? 1.0f : 0.0f;
    v2f B; B.x = lo ? 1.0f : 0.0f;  B.y = lo ? -pc  : 0.0f;
    v8f C = {};
    // d[i][j] = p_i - p_j for this 16x16 subtile
    v8f D = __builtin_amdgcn_wmma_f32_16x16x4_f32(
        false, A, false, B, (short)0, C, false, false);

    float tj = t_col_s[wj * 16 + l15];
    int rbase = wi * 16 + (lo ? 0 : 8);           // 8 consecutive row-targets
    float4 tr0 = *(const float4*)&t_row_s[rbase];
    float4 tr1 = *(const float4*)&t_row_s[rbase + 4];
    float tr[8] = {tr0.x, tr0.y, tr0.z, tr0.w, tr1.x, tr1.y, tr1.z, tr1.w};

    int gj  = bj * 128 + wj * 16 + l15;
    int gi0 = bi * 128 + wi * 16 + (lo ? 0 : 8);

    #pragma unroll
    for (int v = 0; v < 8; ++v) {
      float d  = D[v];
      float a  = fabsf(d);
      // pair-sum over both orderings:
      //   softplus(d)+softplus(-d) - sign(ti-tj)*d = a + 2*ln(1+e^-a) - sgn*d
      float e  = __builtin_amdgcn_exp2f(-a * L2E);   // v_exp_f32
      float sp = __log2f(1.0f + e) * LN2;            // softplus(-a), arg in [1,2]
      float dt = tr[v] - tj;
      float sg = (dt > 0.0f) ? 1.0f : ((dt < 0.0f) ? -1.0f : 0.0f);
      float P  = a + 2.0f * sp - sg * d;
      acc += ((gi0 + v) < gj) ? P : 0.0f;     // strict upper triangle only
    }
  }

  // Deterministic reduction: fixed-order wave shuffle tree, then thread 0.
  float wsum = acc;
  #pragma unroll
  for (int off = 16; off > 0; off >>= 1)
    wsum += __shfl_xor(wsum, off, 32);
  if (lane == 0) wsum_s[wave] = wsum;
  __syncthreads();
  if (t == 0) {
    double s2 = 0.0;
    for (int k = 0; k < 8; ++k) s2 += (double)wsum_s[k];
    block_part[blockIdx.x] = s2;
  }
}

__global__ __launch_bounds__(256) void ranknet_reduce_kernel(
    const double* __restrict__ part, int npart,
    float* __restrict__ out, double inv_count)
{
  __shared__ double s_s[256];
  int t = threadIdx.x;
  double s = 0.0;
  for (int k = t; k < npart; k += 256) s += part[k];
  s_s[t] = s;
  __syncthreads();
  if (t == 0) {
    double tot = 0.0;
    for (int k = 0; k < 256; ++k) tot += s_s[k];
    out[0] = (float)(tot * inv_count);
  }
}

extern "C" void kernel_launch(void* const* d_in, const int* in_sizes, int n_in,
                              void* d_out, int out_size, void* d_ws, size_t ws_size,
                              hipStream_t stream) {
  const float* preds   = (const float*)d_in[0];
  const float* targets = (const float*)d_in[1];
  float* out = (float*)d_out;
  int n  = in_sizes[0];        // 8192
  int NB = n / 128;            // 64 blocks per dim
  int nblocks = NB * (NB + 1) / 2;   // 2080 upper-triangular blocks
  double* part = (double*)d_ws;      // 2080 doubles of scratch

  ranknet_tile_kernel<<<nblocks, 256, 0, stream>>>(preds, targets, part, NB);

  double inv_count = 1.0 / ((double)n * (double)(n - 1));
  ranknet_reduce_kernel<<<1, 256, 0, stream>>>(part, nblocks, out, inv_count);
}